// GCN_88381837017177
// MI455X (gfx1250) — compile-verified
//
#include <hip/hip_runtime.h>
#include <hip/hip_bf16.h>

typedef __attribute__((ext_vector_type(2))) float v2f;
typedef __attribute__((ext_vector_type(8))) float v8f;

#define TPB 256

// ---------------------------------------------------------------------------
// Init: degree starts at 1.0 (self-loop contribution), zero the accumulators.
// ---------------------------------------------------------------------------
__global__ void k_init(float* __restrict__ deg, float* __restrict__ acc, int N) {
    int i = blockIdx.x * blockDim.x + threadIdx.x;
    if (i < N) deg[i] = 1.0f;
    if (i < 8) acc[i] = 0.0f;
}

// ---------------------------------------------------------------------------
// Degree of (A + I) counted at targets.
// ---------------------------------------------------------------------------
__global__ void k_degree(const long long* __restrict__ ei, float* __restrict__ deg,
                         long long E) {
    long long e = (long long)blockIdx.x * blockDim.x + threadIdx.x;
    if (e < E) {
        int dst = (int)ei[E + e];
        atomicAdd(&deg[dst], 1.0f);
    }
}

// ---------------------------------------------------------------------------
// dinv = rsqrt(deg) (in place), and seed s with the self-loop message:
// s[v] = dinv[v]^2 * x[v].
// ---------------------------------------------------------------------------
__global__ void k_dinv_selfloop(const float* __restrict__ x, float* __restrict__ deg_dinv,
                                float* __restrict__ s, int N) {
    int v = blockIdx.x * blockDim.x + threadIdx.x;
    if (v < N) {
        float dv = rsqrtf(deg_dinv[v]);  // deg >= 1 always (self-loop)
        deg_dinv[v] = dv;
        s[v] = dv * dv * x[v];
    }
}

// ---------------------------------------------------------------------------
// Layer-1 scatter (rank-1 collapse): s[dst] += dinv[src]*dinv[dst]*x[src]
// ---------------------------------------------------------------------------
__global__ void k_scatter1(const long long* __restrict__ ei, const float* __restrict__ x,
                           const float* __restrict__ dinv, float* __restrict__ s,
                           long long E) {
    long long e = (long long)blockIdx.x * blockDim.x + threadIdx.x;
    if (e < E) {
        int src = (int)ei[e];
        int dst = (int)ei[E + e];
        float norm = dinv[src] * dinv[dst];
        atomicAdd(&s[dst], norm * x[src]);
    }
}

// ---------------------------------------------------------------------------
// Fused dense transform via V_WMMA_F32_16X16X4_F32 (full fp32 matrix pipe):
//   h1[v,j] = relu(s[v]*W1[j] + b1[j])          (generated in-register)
//   t[v,k]  = sum_j h1[v,j] * W2[j,k]           (WMMA, 16 steps of K=4)
//   out2[v,k] = dinv[v]^2 * t[v,k] + b2[k]      (self-loop seed for layer 2)
// One wave handles a 16-node tile. B (W2 zero-padded to 64x16) is staged in
// LDS TRANSPOSED so each lane's column is contiguous -> plain ds_load_b64,
// no per-iteration EXEC divergence. EXEC is all-ones through the WMMAs.
// ---------------------------------------------------------------------------
__global__ __launch_bounds__(TPB)
void k_transform_wmma(const float* __restrict__ s, const float* __restrict__ dinv,
                      const float* __restrict__ W1, const float* __restrict__ b1,
                      const float* __restrict__ W2, const float* __restrict__ b2,
                      float* __restrict__ t, float* __restrict__ out2, int N) {
    __shared__ float sW1[64], sb1[64];
    __shared__ float sW2t[16 * 64];   // padded B, transposed: [col][k]
    if (threadIdx.x < 64) {
        sW1[threadIdx.x] = W1[threadIdx.x];
        sb1[threadIdx.x] = b1[threadIdx.x];
    }
#pragma unroll
    for (int i = threadIdx.x; i < 16 * 64; i += TPB) {
        int colI = i >> 6;          // padded column 0..15
        int k    = i & 63;          // K index 0..63
        sW2t[i] = (colI < 2) ? W2[k * 2 + colI] : 0.0f;
    }
    __syncthreads();

    int lane = threadIdx.x & 31;
    int wave = threadIdx.x >> 5;
    int tile = blockIdx.x * (TPB / 32) + wave;
    int base = tile * 16;

    int m    = lane & 15;              // A-matrix row this lane contributes to
    int node = base + m;
    float sv = (node < N) ? s[node] : 0.0f;

    int khalf = (lane >> 4) << 1;      // lanes 0-15 -> K offsets 0,1 ; 16-31 -> 2,3
    int col   = lane & 15;             // B / D column
    const float* bcol = &sW2t[col * 64];

    v8f acc = {};
#pragma unroll
    for (int c = 0; c < 16; ++c) {
        int k0 = 4 * c + khalf;        // always even -> 8B-aligned pair loads
        // A tile: 16x4 slice of relu(s*W1 + b1), ISA layout (V0: K=k0, V1: K=k0+1)
        float a0 = fmaxf(fmaf(sv, sW1[k0],     sb1[k0]),     0.0f);
        float a1 = fmaxf(fmaf(sv, sW1[k0 + 1], sb1[k0 + 1]), 0.0f);
        // B tile: contiguous column slice of zero-padded W2 (uniform EXEC)
        float bb0 = bcol[k0];
        float bb1 = bcol[k0 + 1];
        v2f A = {a0, a1};
        v2f B = {bb0, bb1};
        acc = __builtin_amdgcn_wmma_f32_16x16x4_f32(
            /*neg_a=*/false, A, /*neg_b=*/false, B,
            /*c_mod=*/(short)0, acc, /*reuse_a=*/false, /*reuse_b=*/false);
    }

    // D layout: VGPR r holds row M=r (lanes 0-15, N=lane) / M=r+8 (lanes 16-31).
    if (col < 2) {
        int mbase  = (lane >= 16) ? 8 : 0;
        float bias = b2[col];
#pragma unroll
        for (int r = 0; r < 8; ++r) {
            int nd = base + mbase + r;
            if (nd < N) {
                float tv = acc[r];
                t[nd * 2 + col] = tv;
                float dv = dinv[nd];
                out2[nd * 2 + col] = dv * dv * tv + bias;  // self-loop + bias
            }
        }
    }
}

// ---------------------------------------------------------------------------
// Layer-2 scatter: out2[dst,k] += dinv[src]*dinv[dst] * t[src,k]
// ---------------------------------------------------------------------------
__global__ void k_scatter2(const long long* __restrict__ ei, const float* __restrict__ t,
                           const float* __restrict__ dinv, float* __restrict__ out2,
                           long long E) {
    long long e = (long long)blockIdx.x * blockDim.x + threadIdx.x;
    if (e < E) {
        int src = (int)ei[e];
        int dst = (int)ei[E + e];
        float norm = dinv[src] * dinv[dst];
        atomicAdd(&out2[2 * dst],     norm * t[2 * src]);
        atomicAdd(&out2[2 * dst + 1], norm * t[2 * src + 1]);
    }
}

// ---------------------------------------------------------------------------
// Per-node log_softmax (2 classes), block-level tree reduction, atomic add.
// ---------------------------------------------------------------------------
__global__ void k_logsoftmax_reduce(const float* __restrict__ out2,
                                    float* __restrict__ acc, int N) {
    __shared__ float sh0[TPB], sh1[TPB];
    int i = blockIdx.x * blockDim.x + threadIdx.x;
    float l0 = 0.0f, l1 = 0.0f;
    if (i < N) {
        float z0 = out2[2 * i], z1 = out2[2 * i + 1];
        float mx  = fmaxf(z0, z1);
        float lse = mx + logf(expf(z0 - mx) + expf(z1 - mx));
        l0 = z0 - lse;
        l1 = z1 - lse;
    }
    sh0[threadIdx.x] = l0;
    sh1[threadIdx.x] = l1;
    __syncthreads();
    for (int o = TPB / 2; o > 0; o >>= 1) {
        if (threadIdx.x < o) {
            sh0[threadIdx.x] += sh0[threadIdx.x + o];
            sh1[threadIdx.x] += sh1[threadIdx.x + o];
        }
        __syncthreads();
    }
    if (threadIdx.x == 0) {
        atomicAdd(&acc[0], sh0[0]);
        atomicAdd(&acc[1], sh1[0]);
    }
}

__global__ void k_final(const float* __restrict__ acc, float* __restrict__ out, int N) {
    int i = threadIdx.x;
    if (i < 2) out[i] = acc[i] / (float)N;
}

// ---------------------------------------------------------------------------
extern "C" void kernel_launch(void* const* d_in, const int* in_sizes, int n_in,
                              void* d_out, int out_size, void* d_ws, size_t ws_size,
                              hipStream_t stream) {
    (void)n_in; (void)out_size; (void)ws_size;

    const float*     x  = (const float*)d_in[0];
    const long long* ei = (const long long*)d_in[1];  // int64 edge_index [2,E]
    const float*     W1 = (const float*)d_in[2];
    const float*     b1 = (const float*)d_in[3];
    const float*     W2 = (const float*)d_in[4];
    const float*     b2 = (const float*)d_in[5];

    int       N = in_sizes[0];
    long long E = (long long)in_sizes[1] / 2;

    // Workspace carve-up (64-element aligned): dinv | s | t[2N] | out2[2N] | acc
    size_t Np   = ((size_t)(N + 63) / 64) * 64;
    float* deg  = (float*)d_ws;          // degree, then dinv in place
    float* s    = deg  + Np;
    float* t    = s    + Np;
    float* out2 = t    + 2 * Np;
    float* acc  = out2 + 2 * Np;

    int gN = (N + TPB - 1) / TPB;
    int gE = (int)((E + TPB - 1) / TPB);

    k_init<<<gN, TPB, 0, stream>>>(deg, acc, N);
    k_degree<<<gE, TPB, 0, stream>>>(ei, deg, E);
    k_dinv_selfloop<<<gN, TPB, 0, stream>>>(x, deg, s, N);
    k_scatter1<<<gE, TPB, 0, stream>>>(ei, x, deg, s, E);

    int ntiles = (N + 15) / 16;
    int gT = (ntiles + (TPB / 32) - 1) / (TPB / 32);
    k_transform_wmma<<<gT, TPB, 0, stream>>>(s, deg, W1, b1, W2, b2, t, out2, N);

    k_scatter2<<<gE, TPB, 0, stream>>>(ei, t, deg, out2, E);
    k_logsoftmax_reduce<<<gN, TPB, 0, stream>>>(out2, acc, N);
    k_final<<<1, 32, 0, stream>>>(acc, (float*)d_out, N);
}